// MSYNGCN_Torch_11038065951573
// MI455X (gfx1250) — compile-verified
//
#include <hip/hip_runtime.h>
#include <hip/hip_bf16.h>

// ---------------------------------------------------------------------------
// MSYNGCN forward for gfx1250 (MI455X). All dense GEMMs run through
// V_WMMA_F32_16X16X4_F32 (exact fp32 matrix math on the WMMA pipe).
// Round 3: structural ping-pong fragment buffers (K-loop unrolled by 2, no
// register rotation) so the fragment loads of step k+4/k+8 are issued as a
// clause before the 8 WMMAs they overlap, instead of load->wait->wmma chains.
// ---------------------------------------------------------------------------

#define NUU 20000
#define NII 8000
#define NN  28000
#define DD  256
#define DOUT 512
#define DHH 128
#define BB  1024
#define SS  32
#define EA_ 896000
#define ES_ 320000
#define EH_ 128000

typedef float v2f __attribute__((ext_vector_type(2)));
typedef float v8f __attribute__((ext_vector_type(8)));

enum { ACT_NONE = 0, ACT_TANH = 1, ACT_RELU = 2, ACT_SIG = 3 };

__device__ __forceinline__ float act_apply(float x, int act) {
  switch (act) {
    case ACT_TANH: return tanhf(x);
    case ACT_RELU: return fmaxf(x, 0.0f);
    case ACT_SIG:  return 1.0f / (1.0f + expf(-x));
    default:       return x;
  }
}

// ---------------------------------------------------------------------------
// WMMA fp32 GEMM: C = act(A(MxK) * B(KxN or NxK if TRANSB) + bias)
// One wave computes a 32(M) x 64(N) tile: 2 A-fragments x 4 B-fragments,
// 8 f32 accumulators. K loop unrolled by 2 with two structurally-distinct
// fragment buffer sets (ping-pong, no inter-buffer copies), so each 6-load
// group is independent of the 8 WMMAs it is interleaved with.
// Grid: (Ncols/64, M/32), block: 32. All dims are multiples of 32/64 and all
// K are multiples of 8 in this network; EXEC is all-ones around every WMMA.
//
// Fragment layouts (ISA 7.12.2, 32-bit):
//   A 16x4:  lane<16 -> M=lane, {v0,v1}={K=0,K=1}; lane>=16 -> {K=2,K=3}
//   B 4x16:  lane<16 -> N=lane, {v0,v1}={K=0,K=1}; lane>=16 -> {K=2,K=3}
//   C 16x16: lane%16 selects N; VGPR v -> M = v + 8*half
// ---------------------------------------------------------------------------
template <int TRANSB>
__global__ void wmma_gemm_f32(const float* __restrict__ A, int lda,
                              const float* __restrict__ Bm, int ldb,
                              const float* __restrict__ bias,
                              float* __restrict__ C, int ldc,
                              int M, int Ncols, int K, int act) {
  const int lane = threadIdx.x & 31;
  const int r    = lane & 15;
  const int h    = lane >> 4;           // 0/1: half-wave selects K pair / M half
  const int m0   = blockIdx.y << 5;     // 32 rows per wave
  const int n0   = blockIdx.x << 6;     // 64 cols per wave

  v8f acc[2][4];
#pragma unroll
  for (int mi = 0; mi < 2; ++mi)
#pragma unroll
    for (int j = 0; j < 4; ++j) {
      v8f z = {0.f, 0.f, 0.f, 0.f, 0.f, 0.f, 0.f, 0.f};
      acc[mi][j] = z;
    }

  const float* Arow0 = A + (size_t)(m0 + r) * lda;
  const float* Arow1 = A + (size_t)(m0 + 16 + r) * lda;

  auto loadA = [&](int k, v2f& a0, v2f& a1) {
    a0.x = Arow0[k + 2 * h + 0];
    a0.y = Arow0[k + 2 * h + 1];
    a1.x = Arow1[k + 2 * h + 0];
    a1.y = Arow1[k + 2 * h + 1];
  };
  auto loadB = [&](int k, v2f* b) {
#pragma unroll
    for (int j = 0; j < 4; ++j) {
      const int n = n0 + 16 * j + r;
      if (TRANSB) {
        const float* bp = Bm + (size_t)n * ldb + k + 2 * h;
        b[j].x = bp[0];
        b[j].y = bp[1];
      } else {
        b[j].x = Bm[(size_t)(k + 2 * h + 0) * ldb + n];
        b[j].y = Bm[(size_t)(k + 2 * h + 1) * ldb + n];
      }
    }
  };

  // Ping-pong fragment buffers: index 0 / 1 alternate structurally in the
  // unrolled loop body -> no register copies for the compiler to collapse.
  v2f a0[2], a1[2], bf[2][4];
  loadA(0, a0[0], a1[0]);
  loadB(0, bf[0]);

  auto mma = [&](int s) {
#pragma unroll
    for (int j = 0; j < 4; ++j) {
      acc[0][j] = __builtin_amdgcn_wmma_f32_16x16x4_f32(
          false, a0[s], false, bf[s][j], (short)0, acc[0][j], false, false);
      acc[1][j] = __builtin_amdgcn_wmma_f32_16x16x4_f32(
          false, a1[s], false, bf[s][j], (short)0, acc[1][j], false, false);
    }
  };

  for (int k = 0; k < K; k += 8) {        // K % 8 == 0 for every GEMM here
    const int k1 = k + 4;                 // k <= K-8  =>  k1 <= K-4 : valid
    loadA(k1, a0[1], a1[1]);
    loadB(k1, bf[1]);
    mma(0);                               // consumes buf0 while buf1 loads fly

    const int k2 = (k + 8 < K) ? (k + 8) : k;  // clamped on last iteration
    loadA(k2, a0[0], a1[0]);
    loadB(k2, bf[0]);
    mma(1);                               // consumes buf1 while buf0 loads fly
  }

#pragma unroll
  for (int mi = 0; mi < 2; ++mi) {
#pragma unroll
    for (int j = 0; j < 4; ++j) {
      const int col = n0 + 16 * j + r;
      const float bv = bias ? bias[col] : 0.0f;
#pragma unroll
      for (int v = 0; v < 8; ++v) {
        const int row = m0 + 16 * mi + v + 8 * h;
        C[(size_t)row * ldc + col] = act_apply(acc[mi][j][v] + bv, act);
      }
    }
  }
}

// ---------------------------------------------------------------------------
// Helpers
// ---------------------------------------------------------------------------
__global__ void fill_zero(float* p, long long n) {
  long long t = blockIdx.x * (long long)blockDim.x + threadIdx.x;
  if (t < n) p[t] = 0.0f;
}

// segment_sum(v[:,None] * X[cols], rows): one thread handles 4 floats of one edge
__global__ void spmm_atomic(const int* __restrict__ rows, const int* __restrict__ cols,
                            const float* __restrict__ vals, const float* __restrict__ X,
                            float* __restrict__ Y, int E, int D) {
  const int per = D / 4;
  long long t = blockIdx.x * (long long)blockDim.x + threadIdx.x;
  long long total = (long long)E * per;
  if (t >= total) return;
  int e = (int)(t / per);
  int d = (int)(t % per) * 4;
  int rr = rows[e], cc = cols[e];
  float v = vals[e];
  const float* xs = X + (size_t)cc * D + d;
  float* ys = Y + (size_t)rr * D + d;
  atomicAdd(ys + 0, v * xs[0]);
  atomicAdd(ys + 1, v * xs[1]);
  atomicAdd(ys + 2, v * xs[2]);
  atomicAdd(ys + 3, v * xs[3]);
}

// row-wise L2 normalize (x / (||x|| + 1e-9)); one wave per row
__global__ void l2norm_rows(const float* __restrict__ X, float* __restrict__ Y, int D) {
  const int row = blockIdx.x;
  const int lane = threadIdx.x;
  const float* xr = X + (size_t)row * D;
  float s = 0.0f;
  for (int d = lane; d < D; d += 32) { float v = xr[d]; s += v * v; }
  for (int o = 16; o > 0; o >>= 1) s += __shfl_xor(s, o, 32);
  const float inv = 1.0f / (sqrtf(s) + 1e-9f);
  float* yr = Y + (size_t)row * D;
  for (int d = lane; d < D; d += 32) yr[d] = xr[d] * inv;
}

__global__ void concat2(const float* __restrict__ A, int da,
                        const float* __restrict__ B, int db,
                        float* __restrict__ C, int M) {
  const int dc = da + db;
  long long t = blockIdx.x * (long long)blockDim.x + threadIdx.x;
  long long total = (long long)M * dc;
  if (t >= total) return;
  int r = (int)(t / dc);
  int c = (int)(t % dc);
  C[t] = (c < da) ? A[(size_t)r * da + c] : B[(size_t)r * db + (c - da)];
}

__global__ void copy_mat(const float* __restrict__ src, int ls,
                         float* __restrict__ dst, int ld, int M, int nc) {
  long long t = blockIdx.x * (long long)blockDim.x + threadIdx.x;
  long long total = (long long)M * nc;
  if (t >= total) return;
  int r = (int)(t / nc);
  int c = (int)(t % nc);
  dst[(size_t)r * ld + c] = src[(size_t)r * ls + c];
}

// naive GEMM for tiny K/N cases (K=5/8/12, N=2/8/12)
__global__ void sgemm_naive(const float* __restrict__ A, int lda,
                            const float* __restrict__ B, int ldb,
                            const float* __restrict__ bias,
                            float* __restrict__ C, int ldc,
                            int M, int Ncols, int K, int act) {
  long long t = blockIdx.x * (long long)blockDim.x + threadIdx.x;
  long long total = (long long)M * Ncols;
  if (t >= total) return;
  int m = (int)(t / Ncols);
  int n = (int)(t % Ncols);
  float s = bias ? bias[n] : 0.0f;
  for (int k = 0; k < K; ++k) s += A[(size_t)m * lda + k] * B[(size_t)k * ldb + n];
  C[(size_t)m * ldc + n] = act_apply(s, act);
}

__global__ void softmax_rows(float* __restrict__ X, int M, int n) {
  int i = blockIdx.x * blockDim.x + threadIdx.x;
  if (i >= M) return;
  float* row = X + (size_t)i * n;
  float m = row[0];
  for (int j = 1; j < n; ++j) m = fmaxf(m, row[j]);
  float s = 0.0f;
  for (int j = 0; j < n; ++j) { float e = expf(row[j] - m); row[j] = e; s += e; }
  float inv = 1.0f / s;
  for (int j = 0; j < n; ++j) row[j] *= inv;
}

__global__ void gate_mix(const float* __restrict__ g, const float* __restrict__ a,
                         const float* __restrict__ b, float* __restrict__ out, long long n) {
  long long t = blockIdx.x * (long long)blockDim.x + threadIdx.x;
  if (t >= n) return;
  float gg = g[t];
  out[t] = gg * a[t] + (1.0f - gg) * b[t];
}

// out[b, 0:dh] = w[b,0]*pe[b,:], out[b, dh:2dh] = w[b,1]*pz[b,:]
__global__ void scale_concat(const float* __restrict__ w, const float* __restrict__ pe,
                             const float* __restrict__ pz, float* __restrict__ out,
                             int M, int dh) {
  long long t = blockIdx.x * (long long)blockDim.x + threadIdx.x;
  long long total = (long long)M * 2 * dh;
  if (t >= total) return;
  int r = (int)(t / (2 * dh));
  int c = (int)(t % (2 * dh));
  out[t] = (c < dh) ? w[r * 2 + 0] * pe[(size_t)r * dh + c]
                    : w[r * 2 + 1] * pz[(size_t)r * dh + (c - dh)];
}

// Attention pool over S=32 gathered rows of U (stride 512). One wave per batch row.
__global__ void attn_pool(const float* __restrict__ U, const int* __restrict__ idx,
                          const float* __restrict__ mask, const float* __restrict__ aw,
                          const float* __restrict__ ab, float* __restrict__ pooled) {
  const int b = blockIdx.x;
  const int lane = threadIdx.x;                 // lane == s (S==32)
  const int row = idx[b * SS + lane];
  const float* ur = U + (size_t)row * DOUT;
  float sc = 0.0f;
  for (int d = 0; d < DOUT; ++d) sc += ur[d] * aw[d];
  sc += ab[0];
  if (!(mask[b * SS + lane] > 0.0f)) sc = -1e9f;
  // wave32 softmax
  float m = sc;
  for (int o = 16; o > 0; o >>= 1) m = fmaxf(m, __shfl_xor(m, o, 32));
  float e = expf(sc - m);
  float ssum = e;
  for (int o = 16; o > 0; o >>= 1) ssum += __shfl_xor(ssum, o, 32);
  const float alpha = e / ssum;
  // pooled: lane handles 16 contiguous columns
  float acc[16];
#pragma unroll
  for (int j = 0; j < 16; ++j) acc[j] = 0.0f;
  for (int s = 0; s < SS; ++s) {
    float a = __shfl(alpha, s, 32);
    int r2 = __shfl(row, s, 32);
    const float* ur2 = U + (size_t)r2 * DOUT + lane * 16;
#pragma unroll
    for (int j = 0; j < 16; ++j) acc[j] += a * ur2[j];
  }
  float* pd = pooled + (size_t)b * DOUT + lane * 16;
#pragma unroll
  for (int j = 0; j < 16; ++j) pd[j] = acc[j];
}

// ---------------------------------------------------------------------------
// Host orchestration
// ---------------------------------------------------------------------------
static inline long long cdiv(long long a, long long b) { return (a + b - 1) / b; }

extern "C" void kernel_launch(void* const* d_in, const int* in_sizes, int n_in,
                              void* d_out, int out_size, void* d_ws, size_t ws_size,
                              hipStream_t stream) {
  (void)in_sizes; (void)n_in; (void)out_size; (void)ws_size;

  int i = 0;
  const int*   A_rows   = (const int*)d_in[i++];
  const int*   A_cols   = (const int*)d_in[i++];
  const float* A_vals   = (const float*)d_in[i++];
  const int*   S_rows   = (const int*)d_in[i++];
  const int*   S_cols   = (const int*)d_in[i++];
  const float* S_vals   = (const float*)d_in[i++];
  const int*   H_rows   = (const int*)d_in[i++];
  const int*   H_cols   = (const int*)d_in[i++];
  const float* H_vals   = (const float*)d_in[i++];
  const int*   sym_idx  = (const int*)d_in[i++];
  const float* sym_mask = (const float*)d_in[i++];
  const float* user_emb = (const float*)d_in[i++];
  const float* item_emb = (const float*)d_in[i++];
  const float* Qu0 = (const float*)d_in[i++];
  const float* Qu1 = (const float*)d_in[i++];
  const float* Qi0 = (const float*)d_in[i++];
  const float* Qi1 = (const float*)d_in[i++];
  const float* WgcU0_W = (const float*)d_in[i++]; const float* WgcU0_b = (const float*)d_in[i++];
  const float* WgcU1_W = (const float*)d_in[i++]; const float* WgcU1_b = (const float*)d_in[i++];
  const float* WgcI0_W = (const float*)d_in[i++]; const float* WgcI0_b = (const float*)d_in[i++];
  const float* WgcI1_W = (const float*)d_in[i++]; const float* WgcI1_b = (const float*)d_in[i++];
  const float* Mu_W = (const float*)d_in[i++]; const float* Mu_b = (const float*)d_in[i++];
  const float* Mi_W = (const float*)d_in[i++]; const float* Mi_b = (const float*)d_in[i++];
  const float* attn_W = (const float*)d_in[i++]; const float* attn_b = (const float*)d_in[i++];
  const float* mlp1_W = (const float*)d_in[i++]; const float* mlp1_b = (const float*)d_in[i++];
  const float* mlp2_W = (const float*)d_in[i++]; const float* mlp2_b = (const float*)d_in[i++];
  const float* scg_W  = (const float*)d_in[i++]; const float* scg_b  = (const float*)d_in[i++];
  const float* Xfl  = (const float*)d_in[i++];
  const float* Xqi  = (const float*)d_in[i++];
  const float* Xmer = (const float*)d_in[i++];
  const float* Wfl  = (const float*)d_in[i++];
  const float* Wqi  = (const float*)d_in[i++];
  const float* Wmer = (const float*)d_in[i++];
  const float* Wtypes_W = (const float*)d_in[i++]; const float* Wtypes_b = (const float*)d_in[i++];
  const float* Wup_W = (const float*)d_in[i++]; const float* Wup_b = (const float*)d_in[i++];
  const float* gH1_W = (const float*)d_in[i++]; const float* gH1_b = (const float*)d_in[i++];
  const float* gH2_W = (const float*)d_in[i++]; const float* gH2_b = (const float*)d_in[i++];
  const float* hE1_W = (const float*)d_in[i++]; const float* hE1_b = (const float*)d_in[i++];
  const float* hE2_W = (const float*)d_in[i++]; const float* hE2_b = (const float*)d_in[i++];
  const float* hZ1_W = (const float*)d_in[i++]; const float* hZ1_b = (const float*)d_in[i++];
  const float* hZ2_W = (const float*)d_in[i++]; const float* hZ2_b = (const float*)d_in[i++];
  const float* BE = (const float*)d_in[i++];
  const float* BZ = (const float*)d_in[i++];
  const float* cg1_W = (const float*)d_in[i++]; const float* cg1_b = (const float*)d_in[i++];
  const float* cg2_W = (const float*)d_in[i++]; const float* cg2_b = (const float*)d_in[i++];
  const float* Wez_W = (const float*)d_in[i++]; const float* Wez_b = (const float*)d_in[i++];

  // -------- workspace layout (~411 MB of fp32 scratch) --------
  size_t off = 0;
  auto alloc = [&](size_t nfloats) -> float* {
    float* p = (float*)((char*)d_ws + off);
    off += ((nfloats * sizeof(float) + 255) / 256) * 256;
    return p;
  };
  float* PRE   = alloc((size_t)NN * DD);
  float* XBUF  = alloc((size_t)NN * DD);
  float* T1    = alloc((size_t)NN * DD);       // spmm accumulator (reused)
  float* MSG   = alloc((size_t)NN * DD);
  float* CATN  = alloc((size_t)NN * 2 * DD);
  float* XN    = alloc((size_t)NN * DD);
  float* U_OUT = alloc((size_t)NUU * 2 * DD);  // [u_g | u_pair]
  float* I_OUT = alloc((size_t)NII * 2 * DD);  // [i_g | i_pair]
  float* X3    = alloc((size_t)NII * 3 * DD);
  float* HT    = alloc((size_t)NII * DD);
  float* HUP   = alloc((size_t)NII * DOUT);
  float* CATIH = alloc((size_t)NII * 2 * DOUT);
  float* G1    = alloc((size_t)NII * DOUT);
  float* GH    = alloc((size_t)NII * DOUT);
  float* E_H   = alloc((size_t)NII * DOUT);
  float* POOLED= alloc((size_t)BB * DOUT);
  float* TB1   = alloc((size_t)BB * DOUT);
  float* E_GCN = alloc((size_t)BB * DOUT);
  float* TE    = alloc((size_t)BB * DHH);
  float* LE    = alloc((size_t)BB * 8);
  float* PE    = alloc((size_t)BB * DHH);
  float* TZ    = alloc((size_t)BB * DHH);
  float* LZ    = alloc((size_t)BB * 12);
  float* PZ    = alloc((size_t)BB * DHH);
  float* CATEZ = alloc((size_t)BB * 2 * DHH);
  float* TC    = alloc((size_t)BB * DHH);
  float* LW    = alloc((size_t)BB * 2);
  float* SCEZ  = alloc((size_t)BB * 2 * DHH);
  float* E_EZ  = alloc((size_t)BB * DOUT);
  float* CATG2 = alloc((size_t)BB * 2 * DOUT);
  float* G2    = alloc((size_t)BB * DOUT);
  float* E_SC  = alloc((size_t)BB * DOUT);

  auto gemm = [&](const float* A, int lda, const float* Bm, int ldb, const float* bias,
                  float* C, int ldc, int M, int Ncols, int K, int transB, int act) {
    dim3 grid((unsigned)(Ncols / 64), (unsigned)(M / 32));
    if (transB)
      wmma_gemm_f32<1><<<grid, 32, 0, stream>>>(A, lda, Bm, ldb, bias, C, ldc, M, Ncols, K, act);
    else
      wmma_gemm_f32<0><<<grid, 32, 0, stream>>>(A, lda, Bm, ldb, bias, C, ldc, M, Ncols, K, act);
  };
  auto zero = [&](float* p, long long n) {
    fill_zero<<<(unsigned)cdiv(n, 256), 256, 0, stream>>>(p, n);
  };
  auto ew = [&](long long n) { return (unsigned)cdiv(n, 256); };

  // pre = concat(user_emb, item_emb)
  hipMemcpyAsync(PRE, user_emb, (size_t)NUU * DD * sizeof(float), hipMemcpyDeviceToDevice, stream);
  hipMemcpyAsync(PRE + (size_t)NUU * DD, item_emb, (size_t)NII * DD * sizeof(float),
                 hipMemcpyDeviceToDevice, stream);

  // ---- GCN(Qs, Ws, bs): two layers over the full graph ----
  auto run_gcn = [&](const float* Q0, const float* Q1,
                     const float* W0, const float* b0,
                     const float* W1, const float* b1) {
    hipMemcpyAsync(XBUF, PRE, (size_t)NN * DD * sizeof(float), hipMemcpyDeviceToDevice, stream);
    const float* Qs[2] = {Q0, Q1};
    const float* Ws[2] = {W0, W1};
    const float* bs[2] = {b0, b1};
    for (int l = 0; l < 2; ++l) {
      zero(T1, (long long)NN * DD);
      spmm_atomic<<<ew((long long)EA_ * (DD / 4)), 256, 0, stream>>>(
          A_rows, A_cols, A_vals, XBUF, T1, EA_, DD);
      gemm(T1, DD, Qs[l], DD, nullptr, MSG, DD, NN, DD, DD, 0, ACT_TANH);        // msg = tanh(spmm @ Q)
      concat2<<<ew((long long)NN * 2 * DD), 256, 0, stream>>>(XBUF, DD, MSG, DD, CATN, NN);
      gemm(CATN, 2 * DD, Ws[l], DD, bs[l], XN, DD, NN, DD, 2 * DD, 0, ACT_TANH); // x = tanh([x|msg]@W+b)
      l2norm_rows<<<NN, 32, 0, stream>>>(XN, XBUF, DD);                          // x = l2n(x)
    }
  };

  // user-path GCN -> u_g
  run_gcn(Qu0, Qu1, WgcU0_W, WgcU0_b, WgcU1_W, WgcU1_b);
  copy_mat<<<ew((long long)NUU * DD), 256, 0, stream>>>(XBUF, DD, U_OUT, 2 * DD, NUU, DD);
  // item-path GCN -> i_g (rows NU..N-1)
  run_gcn(Qi0, Qi1, WgcI0_W, WgcI0_b, WgcI1_W, WgcI1_b);
  copy_mat<<<ew((long long)NII * DD), 256, 0, stream>>>(XBUF + (size_t)NUU * DD, DD,
                                                        I_OUT, 2 * DD, NII, DD);

  // u_pair = tanh(spmm(S, user_emb) @ Mu_W + Mu_b) -> U_OUT[:,256:512]
  zero(T1, (long long)NUU * DD);
  spmm_atomic<<<ew((long long)ES_ * (DD / 4)), 256, 0, stream>>>(S_rows, S_cols, S_vals,
                                                                 user_emb, T1, ES_, DD);
  gemm(T1, DD, Mu_W, DD, Mu_b, U_OUT + DD, 2 * DD, NUU, DD, DD, 0, ACT_TANH);

  // i_pair = tanh(spmm(H, item_emb) @ Mi_W + Mi_b) -> I_OUT[:,256:512]
  zero(T1, (long long)NII * DD);
  spmm_atomic<<<ew((long long)EH_ * (DD / 4)), 256, 0, stream>>>(H_rows, H_cols, H_vals,
                                                                 item_emb, T1, EH_, DD);
  gemm(T1, DD, Mi_W, DD, Mi_b, I_OUT + DD, 2 * DD, NII, DD, DD, 0, ACT_TANH);

  // ---- item feature branch ----
  sgemm_naive<<<ew((long long)NII * DD), 256, 0, stream>>>(Xfl, 5, Wfl, DD, nullptr,
                                                           X3 + 0, 3 * DD, NII, DD, 5, ACT_NONE);
  sgemm_naive<<<ew((long long)NII * DD), 256, 0, stream>>>(Xqi, 5, Wqi, DD, nullptr,
                                                           X3 + DD, 3 * DD, NII, DD, 5, ACT_NONE);
  sgemm_naive<<<ew((long long)NII * DD), 256, 0, stream>>>(Xmer, 12, Wmer, DD, nullptr,
                                                           X3 + 2 * DD, 3 * DD, NII, DD, 12, ACT_NONE);
  gemm(X3, 3 * DD, Wtypes_W, DD, Wtypes_b, HT, DD, NII, DD, 3 * DD, 0, ACT_NONE);
  gemm(HT, DD, Wup_W, DOUT, Wup_b, HUP, DOUT, NII, DOUT, DD, 0, ACT_NONE);
  concat2<<<ew((long long)NII * 2 * DOUT), 256, 0, stream>>>(I_OUT, DOUT, HUP, DOUT, CATIH, NII);
  gemm(CATIH, 2 * DOUT, gH1_W, DOUT, gH1_b, G1, DOUT, NII, DOUT, 2 * DOUT, 0, ACT_RELU);
  gemm(G1, DOUT, gH2_W, DOUT, gH2_b, GH, DOUT, NII, DOUT, DOUT, 0, ACT_SIG);
  gate_mix<<<ew((long long)NII * DOUT), 256, 0, stream>>>(GH, I_OUT, HUP, E_H, (long long)NII * DOUT);

  // ---- attention pooling + user MLP ----
  attn_pool<<<BB, 32, 0, stream>>>(U_OUT, sym_idx, sym_mask, attn_W, attn_b, POOLED);
  gemm(POOLED, DOUT, mlp1_W, DOUT, mlp1_b, TB1, DOUT, BB, DOUT, DOUT, 0, ACT_RELU);
  gemm(TB1, DOUT, mlp2_W, DOUT, mlp2_b, E_GCN, DOUT, BB, DOUT, DOUT, 0, ACT_NONE);

  // ---- pE / pZ prototype heads ----
  gemm(E_GCN, DOUT, hE1_W, DHH, hE1_b, TE, DHH, BB, DHH, DOUT, 0, ACT_RELU);
  sgemm_naive<<<ew((long long)BB * 8), 256, 0, stream>>>(TE, DHH, hE2_W, 8, hE2_b, LE, 8, BB, 8, DHH, ACT_NONE);
  softmax_rows<<<ew(BB), 256, 0, stream>>>(LE, BB, 8);
  sgemm_naive<<<ew((long long)BB * DHH), 256, 0, stream>>>(LE, 8, BE, DHH, nullptr, PE, DHH, BB, DHH, 8, ACT_NONE);

  gemm(E_GCN, DOUT, hZ1_W, DHH, hZ1_b, TZ, DHH, BB, DHH, DOUT, 0, ACT_RELU);
  sgemm_naive<<<ew((long long)BB * 12), 256, 0, stream>>>(TZ, DHH, hZ2_W, 12, hZ2_b, LZ, 12, BB, 12, DHH, ACT_NONE);
  softmax_rows<<<ew(BB), 256, 0, stream>>>(LZ, BB, 12);
  sgemm_naive<<<ew((long long)BB * DHH), 256, 0, stream>>>(LZ, 12, BZ, DHH, nullptr, PZ, DHH, BB, DHH, 12, ACT_NONE);

  // ---- combine gate ----
  concat2<<<ew((long long)BB * 2 * DHH), 256, 0, stream>>>(PE, DHH, PZ, DHH, CATEZ, BB);
  gemm(CATEZ, 2 * DHH, cg1_W, DHH, cg1_b, TC, DHH, BB, DHH, 2 * DHH, 0, ACT_RELU);
  sgemm_naive<<<ew((long long)BB * 2), 256, 0, stream>>>(TC, DHH, cg2_W, 2, cg2_b, LW, 2, BB, 2, DHH, ACT_NONE);
  softmax_rows<<<ew(BB), 256, 0, stream>>>(LW, BB, 2);
  scale_concat<<<ew((long long)BB * 2 * DHH), 256, 0, stream>>>(LW, PE, PZ, SCEZ, BB, DHH);
  gemm(SCEZ, 2 * DHH, Wez_W, DOUT, Wez_b, E_EZ, DOUT, BB, DOUT, 2 * DHH, 0, ACT_NONE);

  concat2<<<ew((long long)BB * 2 * DOUT), 256, 0, stream>>>(E_GCN, DOUT, E_EZ, DOUT, CATG2, BB);
  gemm(CATG2, 2 * DOUT, scg_W, DOUT, scg_b, G2, DOUT, BB, DOUT, 2 * DOUT, 0, ACT_SIG);
  gate_mix<<<ew((long long)BB * DOUT), 256, 0, stream>>>(G2, E_GCN, E_EZ, E_SC, (long long)BB * DOUT);

  // ---- out = sigmoid(e_sc @ e_H.T) : (1024 x 8000), K=512, B transposed ----
  gemm(E_SC, DOUT, E_H, DOUT, nullptr, (float*)d_out, NII, BB, NII, DOUT, /*transB=*/1, ACT_SIG);
}